// STDG_CGRU_16509854286698
// MI455X (gfx1250) — compile-verified
//
#include <hip/hip_runtime.h>
#include <cstdint>

typedef __attribute__((ext_vector_type(2))) float v2f;
typedef __attribute__((ext_vector_type(8))) float v8f;

#define NN   1024
#define CIN  32
#define TT   12
#define KCH  3
#define HH   64
#define JDIM (CIN*TT)   /* 384 */

static __device__ __forceinline__ v8f wmma_f32(v2f a, v2f b, v8f c) {
  // D = A(16x4) x B(4x16) + C, f32, wave32.
  return __builtin_amdgcn_wmma_f32_16x16x4_f32(false, a, false, b, (short)0, c, false, false);
}

// ---------------- Kernel 1: fused (cheb*STDG)^T @ X  then  @ theta ----------------
// LDS layout (floats):
//   sTheta [96*68]              (padded stride 68)
//   sA     [3*16*16]
//   sX0/1  [16*388] x2          (padded stride 388, double-buffered, async-filled)
//   sC     [3*16*385]           (padded stride 385)
#define K1_THSTR 68
#define K1_XSTR  388
#define K1_XBUF  (16*K1_XSTR)                   /* 6208 floats per X buffer */
#define K1_CSTR  385
#define K1_THETA 0
#define K1_A     (96*K1_THSTR)                  /* 6528 */
#define K1_X0    (K1_A + 3*16*16)               /* 7296 */
#define K1_C     (K1_X0 + 2*K1_XBUF)            /* 19712 */
#define K1_TOTAL (K1_C + 3*16*K1_CSTR)          /* 38192 floats = 152768 B */

__global__ void __launch_bounds__(256)
stdg_cheb_gemm(const float* __restrict__ x, const float* __restrict__ STDG,
               const float* __restrict__ cheb, const float* __restrict__ theta,
               float* __restrict__ gout)
{
  extern __shared__ float smem[];
  float* sTheta = smem + K1_THETA;
  float* sA     = smem + K1_A;
  float* sC     = smem + K1_C;
  // Raw LDS byte offset of the dynamic-shared base (no generic-pointer casts:
  // those form addrspacecast constant expressions lld cannot relocate).
  const uint32_t ldsDyn = (uint32_t)__builtin_amdgcn_groupstaticsize();

  const int tid  = threadIdx.x;
  const int lane = tid & 31;
  const int w    = tid >> 5;
  const int m0   = blockIdx.x * 16;
  const int b    = blockIdx.y;
  const int lm   = lane & 15;
  const int kh   = (lane >> 4) << 1;   // K sub-offset: 0 (lanes 0-15) or 2 (lanes 16-31)
  const int mb   = (lane >> 4) << 3;   // C/D row base per half-wave

  // theta (96x64) -> padded LDS, once
  for (int e = tid; e < 96*64; e += 256) {
    int kc = e >> 6, o = e & 63;
    sTheta[kc*K1_THSTR + o] = theta[e];
  }

  v8f accs[3][3];
#pragma unroll
  for (int kk = 0; kk < 3; ++kk)
#pragma unroll
    for (int jj = 0; jj < 3; ++jj)
      accs[kk][jj] = (v8f){};

  const int jb0 = w * 48;  // this wave owns j columns [jb0, jb0+48)
  const float* Xb = x    + (size_t)b * NN * JDIM;
  const float* Sb = STDG + (size_t)b * NN * NN;

  // async-copy one 16x384 X chunk (n0 = row base) into LDS X buffer `buf`.
  // 1536 float4 / 256 threads -> each wave issues exactly 6 async ops.
  auto issue_x = [&](int n0, int buf) {
    const uint32_t ldsBase = ldsDyn + (uint32_t)((K1_X0 + buf * K1_XBUF) << 2);
    for (int e = tid; e < 1536; e += 256) {
      int n = e / 96, q = (e % 96) << 2;
      uint32_t ldsa = ldsBase + (uint32_t)((n * K1_XSTR + q) << 2);
      uint32_t goff = (uint32_t)((((n0 + n) * JDIM) + q) << 2);
      asm volatile("global_load_async_to_lds_b128 %0, %1, %2 offset:0"
                   :: "v"(ldsa), "v"(goff), "s"(Xb) : "memory");
    }
  };

  // A-product staging: fixed per-thread tile coords (threads 0..191)
  const int aK  = tid / 64, aR = tid % 64;
  const int aN  = aR >> 2, aMq = (aR & 3) << 2;
  const float* chebP = cheb + (size_t)aK*NN*NN + (size_t)aN*NN + m0 + aMq;
  const float* stdgP = Sb + (size_t)aN*NN + m0 + aMq;

  float4 rc, rs;                       // register prefetch for cheb/STDG
  if (tid < 192) { rc = *(const float4*)chebP; rs = *(const float4*)stdgP; }
  issue_x(0, 0);                       // prefetch chunk 0

  for (int nc = 0; nc < NN/16; ++nc) {
    const int cur = nc & 1;
    __syncthreads();                   // chunk nc-1 fully consumed
    // stage A[k][n][m] = cheb * STDG for chunk nc (from prefetched registers)
    if (tid < 192) {
      float4 p; p.x = rc.x*rs.x; p.y = rc.y*rs.y; p.z = rc.z*rs.z; p.w = rc.w*rs.w;
      *(float4*)(sA + aK*256 + aN*16 + aMq) = p;
    }
    if (nc + 1 < NN/16) {
      issue_x((nc + 1) * 16, cur ^ 1);                  // async prefetch next chunk
      if (tid < 192) {                                  // register prefetch next A tiles
        rc = *(const float4*)(chebP + (size_t)(nc + 1) * 16 * NN);
        rs = *(const float4*)(stdgP + (size_t)(nc + 1) * 16 * NN);
      }
      asm volatile("s_wait_asynccnt 6" ::: "memory");   // chunk nc landed; nc+1 in flight
    } else {
      asm volatile("s_wait_asynccnt 0" ::: "memory");
    }
    __syncthreads();

    const float* sX = smem + K1_X0 + cur * K1_XBUF;
#pragma unroll
    for (int ks = 0; ks < 4; ++ks) {
      const int kb = (ks << 2) + kh;
      v2f bf[3];
#pragma unroll
      for (int jj = 0; jj < 3; ++jj) {
        bf[jj].x = sX[kb    *K1_XSTR + jb0 + jj*16 + lm];
        bf[jj].y = sX[(kb+1)*K1_XSTR + jb0 + jj*16 + lm];
      }
#pragma unroll
      for (int kk = 0; kk < 3; ++kk) {
        v2f af;
        af.x = sA[kk*256 + kb*16     + lm];
        af.y = sA[kk*256 + (kb+1)*16 + lm];
#pragma unroll
        for (int jj = 0; jj < 3; ++jj)
          accs[kk][jj] = wmma_f32(af, bf[jj], accs[kk][jj]);
      }
    }
  }

  // dump C tiles (rhs[b,k,m,j]) to LDS for the theta contraction
  __syncthreads();
#pragma unroll
  for (int kk = 0; kk < 3; ++kk)
#pragma unroll
    for (int jj = 0; jj < 3; ++jj)
#pragma unroll
      for (int v = 0; v < 8; ++v)
        sC[kk*16*K1_CSTR + (mb+v)*K1_CSTR + jb0 + jj*16 + lm] = accs[kk][jj][v];
  __syncthreads();

  // out[t][m][o] = sum_{k,c} rhs[k][m][c*12+t] * theta[k][c][o]   (K-dim = 96)
  for (int item = w; item < 48; item += 8) {       // 12 t * 4 o-tiles, wave-uniform
    const int t  = item >> 2;
    const int ob = (item & 3) << 4;
    v8f acc = (v8f){};
    for (int s = 0; s < 24; ++s) {
      const int kc = (s << 2) + kh;
      v2f af, bf;
      { int k1 = kc >> 5, c1 = kc & 31;
        af.x = sC[k1*16*K1_CSTR + lm*K1_CSTR + c1*TT + t]; }
      { int kc2 = kc + 1, k2 = kc2 >> 5, c2 = kc2 & 31;
        af.y = sC[k2*16*K1_CSTR + lm*K1_CSTR + c2*TT + t]; }
      bf.x = sTheta[kc    *K1_THSTR + ob + lm];
      bf.y = sTheta[(kc+1)*K1_THSTR + ob + lm];
      acc = wmma_f32(af, bf, acc);
    }
    float* dst = gout + ((size_t)(b*TT + t)*NN + m0 + mb)*HH + ob + lm;
#pragma unroll
    for (int v = 0; v < 8; ++v)
      dst[(size_t)v*HH] = acc[v];
  }
}

// ---------------- Kernel 2: GRU scan over t, WMMA per 16-row tile ----------------
// LDS layout (floats):
//   sW1T [128*130]  W1 transposed, padded
//   sW2  [128*68]   W2 as-is, padded
//   sH   [128*64]   hidden state
//   sComb[8*16*132] per-wave [g | h] staging, padded
#define K2_W1STR 130
#define K2_W2STR 68
#define K2_CSTR  132
#define K2_W1T   0
#define K2_W2    (128*K2_W1STR)                 /* 16640 */
#define K2_H     (K2_W2 + 128*K2_W2STR)        /* 25344 */
#define K2_COMB  (K2_H + 128*HH)               /* 33536 */
#define K2_TOTAL (K2_COMB + 8*16*K2_CSTR)      /* 50432 floats */

__global__ void __launch_bounds__(256)
gru_scan(const float* __restrict__ W1, const float* __restrict__ b1,
         const float* __restrict__ W2, const float* __restrict__ b2,
         const float* __restrict__ conv_w, const float* __restrict__ conv_b,
         float* __restrict__ gio)
{
  extern __shared__ float smem[];
  float* sW1T  = smem + K2_W1T;
  float* sW2   = smem + K2_W2;
  float* sH    = smem + K2_H;
  float* sComb = smem + K2_COMB;

  const int tid  = threadIdx.x;
  const int lane = tid & 31;
  const int w    = tid >> 5;
  const int lm   = lane & 15;
  const int kh   = (lane >> 4) << 1;
  const int mb   = (lane >> 4) << 3;
  const int wrow = blockIdx.x * 128 + w * 16;   // first of this wave's 16 rows

  for (int e = tid; e < 128*128; e += 256) {    // sW1T[j][i] = W1[i][j]
    int i = e >> 7, j = e & 127;
    sW1T[j*K2_W1STR + i] = W1[e];
  }
  for (int e = tid; e < 128*64; e += 256) {     // sW2[j][o] = W2[j][o]
    int j = e >> 6, o = e & 63;
    sW2[j*K2_W2STR + o] = W2[e];
  }
  for (int e = tid; e < 128*HH; e += 256) sH[e] = 0.0f;  // h0 = 0
  __syncthreads();

  float* combW = sComb + w*16*K2_CSTR;
  float* hW    = sH + w*16*HH;

  for (int t = 0; t < TT; ++t) {
    // stage comb = [ g(t) | h ]   (16 rows x 32 float4)
    for (int e = lane; e < 512; e += 32) {
      int m = e >> 5, q = e & 31;
      if (q < 16) {
        int row = wrow + m, bb = row >> 10, n = row & 1023;
        float4 g4 = *(const float4*)(gio + ((size_t)(bb*TT + t)*NN + n)*HH + (q << 2));
        *(float4*)(combW + m*K2_CSTR + (q << 2)) = g4;
      } else {
        float4 h4 = *(const float4*)(hW + m*HH + ((q - 16) << 2));
        *(float4*)(combW + m*K2_CSTR + 64 + ((q - 16) << 2)) = h4;
      }
    }
    __syncthreads();

    // zr = comb @ W1^T + b1   (16 x 128)
    v8f zacc[8];
#pragma unroll
    for (int ot = 0; ot < 8; ++ot) {
      float bias = b1[(ot << 4) + lm];
      v8f acc;
#pragma unroll
      for (int v = 0; v < 8; ++v) acc[v] = bias;
      for (int s = 0; s < 32; ++s) {
        int k = (s << 2) + kh;
        v2f af, bf;
        af.x = combW[lm*K2_CSTR + k];
        af.y = combW[lm*K2_CSTR + k + 1];
        bf.x = sW1T[k    *K2_W1STR + (ot << 4) + lm];
        bf.y = sW1T[(k+1)*K2_W1STR + (ot << 4) + lm];
        acc = wmma_f32(af, bf, acc);
      }
      zacc[ot] = acc;
    }

    // comb[:,64:128] = sigmoid(r) * h   (r = zr cols 64..127)
#pragma unroll
    for (int ot = 4; ot < 8; ++ot)
#pragma unroll
      for (int v = 0; v < 8; ++v) {
        int m = mb + v, o = ((ot - 4) << 4) + lm;
        float r = 1.0f / (1.0f + __expf(-zacc[ot][v]));
        combW[m*K2_CSTR + 64 + o] = r * hW[m*HH + o];
      }
    __syncthreads();

    // cand = [g | r*h] @ W2 + b2   (16 x 64)
    v8f cacc[4];
#pragma unroll
    for (int ot = 0; ot < 4; ++ot) {
      float bias = b2[(ot << 4) + lm];
      v8f acc;
#pragma unroll
      for (int v = 0; v < 8; ++v) acc[v] = bias;
      for (int s = 0; s < 32; ++s) {
        int k = (s << 2) + kh;
        v2f af, bf;
        af.x = combW[lm*K2_CSTR + k];
        af.y = combW[lm*K2_CSTR + k + 1];
        bf.x = sW2[k    *K2_W2STR + (ot << 4) + lm];
        bf.y = sW2[(k+1)*K2_W2STR + (ot << 4) + lm];
        acc = wmma_f32(af, bf, acc);
      }
      cacc[ot] = acc;
    }

    // h = z*h + (1-z)*tanh(cand)
#pragma unroll
    for (int ot = 0; ot < 4; ++ot)
#pragma unroll
      for (int v = 0; v < 8; ++v) {
        int m = mb + v, o = (ot << 4) + lm;
        float z = 1.0f / (1.0f + __expf(-zacc[ot][v]));
        float cd = tanhf(cacc[ot][v]);
        float hold = hW[m*HH + o];
        hW[m*HH + o] = z*hold + (1.0f - z)*cd;
      }
    __syncthreads();
  }

  // out[b,t,n,o] = h[n,o]*conv_w[t] + conv_b[t]
  for (int t = 0; t < TT; ++t) {
    float wt = conv_w[t], cb = conv_b[t];
    for (int e = lane; e < 256; e += 32) {
      int m = e >> 4, q = (e & 15) << 2;
      int row = wrow + m, bb = row >> 10, n = row & 1023;
      float4 h4 = *(const float4*)(hW + m*HH + q);
      float4 o4; o4.x = h4.x*wt + cb; o4.y = h4.y*wt + cb;
      o4.z = h4.z*wt + cb; o4.w = h4.w*wt + cb;
      *(float4*)(gio + ((size_t)(bb*TT + t)*NN + n)*HH + q) = o4;
    }
  }
}

extern "C" void kernel_launch(void* const* d_in, const int* in_sizes, int n_in,
                              void* d_out, int out_size, void* d_ws, size_t ws_size,
                              hipStream_t stream) {
  (void)in_sizes; (void)n_in; (void)d_ws; (void)ws_size; (void)out_size;
  const float* x      = (const float*)d_in[0];
  const float* STDG   = (const float*)d_in[1];
  const float* cheb   = (const float*)d_in[2];
  const float* theta  = (const float*)d_in[3];
  const float* W1     = (const float*)d_in[4];
  const float* b1     = (const float*)d_in[5];
  const float* W2     = (const float*)d_in[6];
  const float* b2     = (const float*)d_in[7];
  const float* conv_w = (const float*)d_in[8];
  const float* conv_b = (const float*)d_in[9];
  float* out = (float*)d_out;

  // Kernel 1: G[b,t,n,h] written into d_out (same shape as final output)
  dim3 g1(NN/16, 16);
  stdg_cheb_gemm<<<g1, 256, K1_TOTAL * sizeof(float), stream>>>(x, STDG, cheb, theta, out);
  // Kernel 2: GRU scan in-place over d_out, then conv broadcast overwrite
  gru_scan<<<16*NN/128, 256, K2_TOTAL * sizeof(float), stream>>>(W1, b1, W2, b2, conv_w, conv_b, out);
}